// SSFAttBlock_3195455668600
// MI455X (gfx1250) — compile-verified
//
#include <hip/hip_runtime.h>
#include <hip/hip_bf16.h>
#include <math.h>

#define NB        50
#define IMG_W     512
#define IMG_H     512
#define NPIX      (IMG_W * IMG_H)     /* 262144 = 2^18 */
#define NPIX_SH   18
#define BATCH     32
#define THETA_F   30.0f
#define THR_AREA  16.0f               /* bin_size = 512/32 */
#define HTILE     2048                /* floats staged per block in k_hist  */
#define FTILE     2048                /* floats per tile in k_final        */
#define FITER     4                   /* tiles per block in k_final        */

typedef unsigned int u32x4 __attribute__((ext_vector_type(4)));
typedef int          i32x8 __attribute__((ext_vector_type(8)));
typedef int          i32x4 __attribute__((ext_vector_type(4)));

#if defined(__gfx1250__) && __has_builtin(__builtin_amdgcn_tensor_load_to_lds)
#define HAVE_TDM 1
#else
#define HAVE_TDM 0
#endif

// Generic-pointer low 32 bits == LDS byte offset (SHARED aperture mapping).
#define LDS_ADDR(p) ((unsigned)(unsigned long long)(void*)(p))

#if HAVE_TDM
// Issue one TDM DMA: n contiguous dwords  global -> LDS. Tracked by TENSORcnt.
__device__ __forceinline__ void tdm_load_dwords(unsigned lds_byte,
                                                unsigned long long ga, int n) {
    u32x4 g0;
    g0[0] = 1u;                                       // count=1, user descriptor
    g0[1] = lds_byte;                                 // lds_addr
    g0[2] = (unsigned)(ga & 0xFFFFFFFFu);             // global_addr[31:0]
    g0[3] = (unsigned)((ga >> 32) & 0x01FFFFFFu) | 0x80000000u; // addr[56:32]|type=2
    i32x8 g1;
    g1[0] = (2 << 16);                                // data_size = 2 (4 bytes)
    g1[1] = (n & 0xFFFF) << 16;                       // tensor_dim0[15:0]
    g1[2] = ((n >> 16) & 0xFFFF) | (1 << 16);         // tensor_dim0[31:16] | tensor_dim1=1
    g1[3] = (n & 0xFFFF) << 16;                       // tile_dim0 = n
    g1[4] = 1;                                        // tile_dim1=1, tile_dim2=0
    g1[5] = n;                                        // tensor_dim0_stride
    g1[6] = (n & 0xFFFF) << 16;                       // tensor_dim1_stride lo16
    g1[7] = 0;
    i32x4 z4 = {0, 0, 0, 0};
    i32x8 z8 = {0, 0, 0, 0, 0, 0, 0, 0};
    __builtin_amdgcn_tensor_load_to_lds(g0, g1, z4, z4, z8, 0);
}
#endif

// ---------------------------------------------------------------- histogram
__global__ void k_hist(const float* __restrict__ x, int* __restrict__ hist) {
    __shared__ float xs[HTILE];
    __shared__ int   sh[NB];
    if (threadIdx.x < NB) sh[threadIdx.x] = 0;
    const long long base = (long long)blockIdx.x * HTILE;   // HTILE | NPIX -> one image
#if HAVE_TDM
    if (threadIdx.x < 32) {                            // wave 0 drives the DMA
        tdm_load_dwords(LDS_ADDR(xs), (unsigned long long)(const void*)(x + base), HTILE);
        __builtin_amdgcn_s_wait_tensorcnt(0);
    }
#else
    for (int i = threadIdx.x; i < HTILE; i += blockDim.x) xs[i] = x[base + i];
#endif
    __syncthreads();
    const int b = (int)(base >> NPIX_SH);
#pragma unroll
    for (int k = 0; k < HTILE / 256; ++k) {
        float v = xs[threadIdx.x + k * 256];
        int bin = (int)(v * (float)NB);
        bin = bin < 0 ? 0 : (bin > NB - 1 ? NB - 1 : bin);
        atomicAdd(&sh[bin], 1);                        // ds_add_u32
    }
    __syncthreads();
    if (threadIdx.x < NB) atomicAdd(&hist[b * NB + threadIdx.x], sh[threadIdx.x]);
}

// ------------------------------------------------- per-image threshold (tiny)
__global__ void k_thr(const int* __restrict__ hist, float* __restrict__ thr) {
    int b = blockIdx.x;
    if (threadIdx.x != 0) return;
    const int* h = hist + b * NB;
    int best = h[0], im = 0;                           // last argmax
    for (int i = 1; i < NB; ++i) if (h[i] >= best) { best = h[i]; im = i; }
    int best2 = -2147483647, is = 0;                   // last argmax of masked hist
    for (int i = 0; i < NB; ++i) {
        int v = (i > im) ? h[i] : -1;
        if (v >= best2) { best2 = v; is = i; }
    }
    thr[b] = (float)is / (float)NB;
}

// ---------------------------------------------------------------- init labels
__global__ void k_init(int* __restrict__ labels) {
    int gp = blockIdx.x * blockDim.x + threadIdx.x;
    labels[gp] = gp & (NPIX - 1);                      // per-image linear index
}

// ------------------------------------------------------- union-find primitives
__device__ __forceinline__ void unite(int* l, int a, int b) {
    while (true) {
        while (l[a] != a) a = l[a];
        while (l[b] != b) b = l[b];
        if (a == b) return;
        if (a > b) { int t = a; a = b; b = t; }        // a < b
        int old = atomicMin(&l[b], a);
        if (old == b) return;
        b = old;                                        // retry from displaced root
    }
}

// ------------------------------------------------------------------- merge
__global__ void k_merge(const float* __restrict__ x, const float* __restrict__ thr,
                        int* __restrict__ labels) {
    int gp = blockIdx.x * blockDim.x + threadIdx.x;
    int b  = gp >> NPIX_SH;
    int p  = gp & (NPIX - 1);
    float t = thr[b];
    const float* xb = x + ((long long)b << NPIX_SH);
    if (xb[p] <= t) return;                            // not in mask
    int* l = labels + ((long long)b << NPIX_SH);
    int row = p >> 9, col = p & (IMG_W - 1);
    // union with the 4 "previous" 8-neighbors covers every adjacent pair once
    if (col > 0             && xb[p - 1]         > t) unite(l, p, p - 1);
    if (row > 0) {
        if (                   xb[p - IMG_W]     > t) unite(l, p, p - IMG_W);
        if (col > 0         && xb[p - IMG_W - 1] > t) unite(l, p, p - IMG_W - 1);
        if (col < IMG_W - 1 && xb[p - IMG_W + 1] > t) unite(l, p, p - IMG_W + 1);
    }
}

// -------------------------------------------------- flatten + component counts
__global__ void k_flatten(const float* __restrict__ x, const float* __restrict__ thr,
                          int* __restrict__ labels, int* __restrict__ counts) {
    int gp = blockIdx.x * blockDim.x + threadIdx.x;
    int b  = gp >> NPIX_SH;
    int p  = gp & (NPIX - 1);
    if (x[gp] <= thr[b]) return;                       // non-mask keeps label == p
    int* l = labels + ((long long)b << NPIX_SH);
    int r = p;
    while (l[r] != r) r = l[r];
    l[p] = r;
    atomicAdd(&counts[((long long)b << NPIX_SH) + r], 1);
}

// -------------------------------------------------------------------- final
// Persistent block over FITER tiles, double-buffered TDM pipeline:
// wave0 issues tile t+1 (x + labels pair), then waits tensorcnt<=2 so the
// in-order TDM guarantees tile t has landed; barriers fence buffer reuse.
// area>1 already implies mask-pixel, so output needs only x, labels, counts.
__global__ void k_final(const float* __restrict__ x, const int* __restrict__ labels,
                        const int* __restrict__ counts, float* __restrict__ out) {
    __shared__ float xs[2][FTILE];
    __shared__ int   ls[2][FTILE];
    const long long blockBase = (long long)blockIdx.x * (FTILE * FITER);
    const int b = (int)(blockBase >> NPIX_SH);         // FTILE*FITER | NPIX
    const int* cb = counts + ((long long)b << NPIX_SH);

#if HAVE_TDM
    if (threadIdx.x < 32) {                            // prologue: tile 0 pair
        tdm_load_dwords(LDS_ADDR(xs[0]), (unsigned long long)(const void*)(x + blockBase), FTILE);
        tdm_load_dwords(LDS_ADDR(ls[0]), (unsigned long long)(const void*)(labels + blockBase), FTILE);
    }
#endif
    for (int t = 0; t < FITER; ++t) {
        const int cur = t & 1;
        const long long tbase = blockBase + (long long)t * FTILE;
#if HAVE_TDM
        if (threadIdx.x < 32) {
            if (t + 1 < FITER) {                       // prefetch next pair
                const long long nbase = tbase + FTILE;
                tdm_load_dwords(LDS_ADDR(xs[cur ^ 1]),
                                (unsigned long long)(const void*)(x + nbase), FTILE);
                tdm_load_dwords(LDS_ADDR(ls[cur ^ 1]),
                                (unsigned long long)(const void*)(labels + nbase), FTILE);
                __builtin_amdgcn_s_wait_tensorcnt(2);  // oldest pair (tile t) done
            } else {
                __builtin_amdgcn_s_wait_tensorcnt(0);
            }
        }
#else
        for (int i = threadIdx.x; i < FTILE; i += blockDim.x) {
            xs[cur][i] = x[tbase + i];
            ls[cur][i] = labels[tbase + i];
        }
#endif
        __syncthreads();                               // tile t visible to all waves
#pragma unroll
        for (int k = 0; k < FTILE / 256; ++k) {
            int li = threadIdx.x + k * 256;
            float xv = xs[cur][li];
            int r = ls[cur][li];
            float area = sqrtf((float)cb[r]);
            float o = xv;
            if (area > 1.0f) {                         // == apply (mask implied)
                float value = (area < THR_AREA) ? 0.5f
                            : 1.0f / (1.0f + expf(-THETA_F * (area - THR_AREA)));
                o = powf(xv, 1.0f / value);
            }
            out[tbase + li] = o;
        }
        __syncthreads();                               // fence before buffer reuse
    }
}

// ---------------------------------------------------------------------------
extern "C" void kernel_launch(void* const* d_in, const int* in_sizes, int n_in,
                              void* d_out, int out_size, void* d_ws, size_t ws_size,
                              hipStream_t stream) {
    const float* x  = (const float*)d_in[0];           // (32,1,512,512) f32
    float* out      = (float*)d_out;                   // same shape
    char* w         = (char*)d_ws;
    // workspace layout (needs ~67.2 MB)
    int*   hist   = (int*)w;                           // BATCH*NB ints
    float* thr    = (float*)(w + 8192);                // BATCH floats
    int*   labels = (int*)(w + 16384);                 // BATCH*NPIX ints (32 MB)
    int*   counts = labels + (size_t)BATCH * NPIX;     // BATCH*NPIX ints (32 MB)

    (void)hipMemsetAsync(hist,   0, (size_t)BATCH * NB * sizeof(int), stream);
    (void)hipMemsetAsync(counts, 0, (size_t)BATCH * NPIX * sizeof(int), stream);

    const int total = BATCH * NPIX;                    // 8388608
    k_hist   <<<total / HTILE, 256, 0, stream>>>(x, hist);
    k_thr    <<<BATCH, 32, 0, stream>>>(hist, thr);
    k_init   <<<total / 256, 256, 0, stream>>>(labels);
    k_merge  <<<total / 256, 256, 0, stream>>>(x, thr, labels);
    k_flatten<<<total / 256, 256, 0, stream>>>(x, thr, labels, counts);
    k_final  <<<total / (FTILE * FITER), 256, 0, stream>>>(x, labels, counts, out);
}